// WaveletAttention_2972117369305
// MI455X (gfx1250) — compile-verified
//
#include <hip/hip_runtime.h>

// WaveletAttention fused Haar DWT -> affine -> IDWT.
// Entire op == per-2x2-block affine map out = T(4x4)*v + bias, executed as
// V_WMMA_F32_16X16X4_F32 with A = 4 stacked copies of T, B = 16 data blocks,
// C = per-component bias. Memory-bound: 512MB @ 23.3TB/s ~ 22us floor.

typedef float v2f __attribute__((ext_vector_type(2)));
typedef float v8f __attribute__((ext_vector_type(8)));

#define TOTAL_TILES (1u << 20)   // 2^26 elems / 64 elems per 16-block tile
#define NBLOCKS 8192
#define NTHREADS 256
#define WAVES_PER_BLOCK (NTHREADS / 32)
#define NWAVES (NBLOCKS * WAVES_PER_BLOCK)       // 65536
#define TILES_PER_WAVE (TOTAL_TILES / NWAVES)    // 16 (uniform => EXEC stays all-1s)

__global__ __launch_bounds__(NTHREADS) void wavelet_haar_wmma(
    const float* __restrict__ x,
    const float* __restrict__ alpha_p, const float* __restrict__ beta_p,
    const float* __restrict__ w1_p,    const float* __restrict__ b1_p,
    const float* __restrict__ w2_p,    const float* __restrict__ b2_p,
    float* __restrict__ out)
{
    // ---- fold all scalars into the 4x4 transform T and bias vector ----
    const float alpha = alpha_p[0], beta = beta_p[0];
    const float w1 = w1_p[0], b1 = b1_p[0], w2 = w2_p[0], b2 = b2_p[0];
    const float P  = alpha * w1;
    const float Q  = beta * w2;
    const float R  = alpha * b1 + 3.0f * beta * b2;
    const float K1 = 0.25f * (P - Q - w2);   // coefficient on s = a+b+c+d
    const float K2 = Q;                      // extra coefficient on 'a'
    const float K3 = 0.5f * R;
    const float bias0 = K3 + 1.5f * b2;      // component 0 (a')
    const float biasX = K3 - 0.5f * b2;      // components 1..3 (b',c',d')

    const int lane = threadIdx.x & 31;
    const int half = lane >> 4;      // 0: top row of block (K=0,1)  1: bottom (K=2,3)
    const int col  = lane & 15;      // block index within tile / D column
    const int j    = lane & 3;       // A-matrix row % 4 == output component
    const int k0   = half << 1;      // first K slice held by this lane

    // A (16x4) = [T;T;T;T], T[j][k] = K1 + K2*d(k==0) + w2*d(j==k)
    // layout: V0 holds K=k0, V1 holds K=k0+1 (lane-half selects K pair)
    v2f Amat;
    Amat.x = K1 + (k0 == 0 ? K2 : 0.0f) + (j == k0       ? w2 : 0.0f);
    Amat.y = K1 +                         (j == (k0 | 1) ? w2 : 0.0f);

    // C (16x16): row m needs bias_{m%4}; C VGPR v covers rows v and v+8,
    // both with the same m%4 -> lane-uniform broadcast per VGPR.
    v8f Cmat;
    Cmat[0] = bias0; Cmat[1] = biasX; Cmat[2] = biasX; Cmat[3] = biasX;
    Cmat[4] = bias0; Cmat[5] = biasX; Cmat[6] = biasX; Cmat[7] = biasX;

    const unsigned wave = blockIdx.x * WAVES_PER_BLOCK + (threadIdx.x >> 5);
    // lane's float2 slot inside a tile: top-row pair or bottom-row pair (W=256)
    const unsigned laneOff = (unsigned)(col << 1) + (unsigned)(half << 8);

    const unsigned t0 = wave * TILES_PER_WAVE;
    #pragma unroll 4
    for (int i = 0; i < TILES_PER_WAVE; ++i) {
        const unsigned t = t0 + i;
        // tile t: row-pair r = t>>3 starts at float offset r*512 (2 rows of 256),
        // w-segment (t&7)*32
        const size_t off = ((size_t)(t >> 3) << 9) + (size_t)((t & 7u) << 5) + laneOff;

        // B (4x16): lane 'col' holds block col's (a,b); lane col+16 holds (c,d)
        v2f Bmat = __builtin_nontemporal_load((const v2f*)(x + off));

        // D = A*B + C : column n = T * block_n + bias
        v8f D = __builtin_amdgcn_wmma_f32_16x16x4_f32(
            false, Amat, false, Bmat, (short)0, Cmat, false, false);

        // every lane holds (a',b',c',d') of block col in D[0..3];
        // top lanes store (a',b'), bottom lanes store (c',d') at the load address
        v2f o;
        o.x = half ? D[2] : D[0];
        o.y = half ? D[3] : D[1];
        __builtin_nontemporal_store(o, (v2f*)(out + off));
    }
}

extern "C" void kernel_launch(void* const* d_in, const int* in_sizes, int n_in,
                              void* d_out, int out_size, void* d_ws, size_t ws_size,
                              hipStream_t stream) {
    (void)in_sizes; (void)n_in; (void)out_size; (void)d_ws; (void)ws_size;
    const float* x     = (const float*)d_in[0];
    const float* alpha = (const float*)d_in[1];
    const float* beta  = (const float*)d_in[2];
    const float* w1    = (const float*)d_in[3];
    const float* b1    = (const float*)d_in[4];
    const float* w2    = (const float*)d_in[5];
    const float* b2    = (const float*)d_in[6];
    float* out = (float*)d_out;

    wavelet_haar_wmma<<<NBLOCKS, NTHREADS, 0, stream>>>(
        x, alpha, beta, w1, b1, w2, b2, out);
}